// TFNLayer_34024730919468
// MI455X (gfx1250) — compile-verified
//
#include <hip/hip_runtime.h>
#include <hip/hip_bf16.h>

typedef float v2f __attribute__((ext_vector_type(2)));
typedef float v8f __attribute__((ext_vector_type(8)));

#define NN 384

// workspace layout (float offsets)
// G tensors are stored in WMMA-fragment-ready order:
//   G??p[j][k][lane][2] where for k-step k (rows h0=4k..4k+3 of the K x 16
//   matrix), lane l holds elements (h = 4k + 2*(l>>4) + e, o = l&15), e=0,1.
#define OFF_G00  0u          // [384][8][32][2]   (512/j)
#define OFF_G01  196608u     // [384][8][32][2]
#define OFF_G10  393216u     // [384][3][8][32][2] (1536/j)
#define OFF_G11  983040u     // [384][3][8][32][2]
#define OFF_FB00 1572864u    // [384][16]
#define OFF_FB01 1579008u    // [384][16]
#define OFF_FB10 1585152u    // [384][3][16]
#define OFF_FB11 1603584u    // [384][3][16]
#define OFF_MA   1622016u    // [24][384][16]
#define OFF_MV   1769472u    // [24][384][3][16]
#define OFF_AP   2211840u    // [4 branches][2 nt][4 k][64][... ] = 4*512 packed A frags

typedef const __attribute__((address_space(1))) v2f* gv2p;
static __device__ inline v2f ldg2(const float* p) {   // coalesced global b64 load
    return *(gv2p)p;
}

static __device__ inline v8f wmma4(v2f a, v2f b, v8f c) {
    return __builtin_amdgcn_wmma_f32_16x16x4_f32(false, a, false, b, (short)0, c,
                                                 false, false);
}

// fragment-slot for matrix element (row h of K, col o), K tiled by 4
static __device__ inline int frag_slot(int h, int o) {
    const int k = h >> 2, e = h & 1, half = (h >> 1) & 1;
    return k * 64 + (half * 16 + o) * 2 + e;
}

// ---------------------------------------------------------------------------
// Kernel 1: pre-contract node features into second-layer MLP weights and
// store in WMMA-fragment order; also pack stage-1 A matrices; fold biases.
// ---------------------------------------------------------------------------
__global__ __launch_bounds__(256) void precompute_kernel(
    const float* __restrict__ f0, const float* __restrict__ f1,
    const float* __restrict__ B00, const float* __restrict__ b00,
    const float* __restrict__ B10, const float* __restrict__ b10,
    const float* __restrict__ B01, const float* __restrict__ b01,
    const float* __restrict__ B11, const float* __restrict__ b11,
    const float* __restrict__ A00, const float* __restrict__ A10,
    const float* __restrict__ A01, const float* __restrict__ A11,
    float* __restrict__ ws)
{
    const int j = blockIdx.x;
    const int tid = threadIdx.x;
    __shared__ float sf0[16], sf1[48];
    if (tid < 16) sf0[tid] = f0[j * 16 + tid];
    if (tid < 48) sf1[tid] = f1[j * 48 + tid];
    __syncthreads();

    float* G00 = ws + OFF_G00 + (size_t)j * 512;
    float* G01 = ws + OFF_G01 + (size_t)j * 512;
    float* G10 = ws + OFF_G10 + (size_t)j * 1536;
    float* G11 = ws + OFF_G11 + (size_t)j * 1536;

    for (int idx = tid; idx < 512; idx += 256) {
        const int h = idx >> 4, o = idx & 15;
        float s0 = 0.f, s1 = 0.f;
        #pragma unroll
        for (int d = 0; d < 16; ++d) {
            const float w = sf0[d];
            s0 += w * B00[h * 256 + d * 16 + o];
            s1 += w * B01[h * 256 + d * 16 + o];
        }
        const int slot = frag_slot(h, o);
        G00[slot] = s0;
        G01[slot] = s1;
    }
    for (int idx = tid; idx < 1536; idx += 256) {
        const int x = idx / 512, rem = idx & 511;
        const int h = rem >> 4, o = rem & 15;
        float s0 = 0.f, s1 = 0.f;
        #pragma unroll
        for (int c = 0; c < 16; ++c) {
            const float w = sf1[c * 3 + x];
            s0 += w * B10[h * 256 + c * 16 + o];
            s1 += w * B11[h * 256 + c * 16 + o];
        }
        const int slot = x * 512 + frag_slot(h, o);
        G10[slot] = s0;
        G11[slot] = s1;
    }
    if (tid < 16) {
        float s0 = 0.f, s1 = 0.f;
        #pragma unroll
        for (int d = 0; d < 16; ++d) {
            s0 += sf0[d] * b00[d * 16 + tid];
            s1 += sf0[d] * b01[d * 16 + tid];
        }
        ws[OFF_FB00 + j * 16 + tid] = s0;
        ws[OFF_FB01 + j * 16 + tid] = s1;
    }
    if (tid < 48) {
        const int x = tid >> 4, o = tid & 15;
        float s0 = 0.f, s1 = 0.f;
        #pragma unroll
        for (int c = 0; c < 16; ++c) {
            s0 += sf1[c * 3 + x] * b10[c * 16 + o];
            s1 += sf1[c * 3 + x] * b11[c * 16 + o];
        }
        ws[OFF_FB10 + j * 48 + tid] = s0;
        ws[OFF_FB11 + j * 48 + tid] = s1;
    }
    // pack stage-1 A matrices [16 x 32] into fragment order (blocks j=0..3)
    if (j < 4) {
        const float* Am = (j == 0) ? A00 : (j == 1) ? A10 : (j == 2) ? A01 : A11;
        for (int idx = tid; idx < 512; idx += 256) {
            const int nt = idx >> 8, rem = idx & 255;
            const int k = rem >> 6, rem2 = rem & 63;
            const int ln = rem2 >> 1, e = rem2 & 1;
            const int row = k * 4 + 2 * (ln >> 4) + e;   // K index (rbf dim)
            const int col = nt * 16 + (ln & 15);         // hidden index
            ws[OFF_AP + j * 512 + idx] = Am[row * 32 + col];
        }
    }
}

// ---------------------------------------------------------------------------
// Kernel 2: per 16i x 16j tile: stage-1 MLP (rbf@A + SiLU) via WMMA, then the
// four factorized contraction GEMMs via WMMA, partial sums to workspace.
// Block = 128 threads = 4 waves (wave32). grid = (24 jt, 24 it).
// ---------------------------------------------------------------------------
__global__ __launch_bounds__(128) void pair_kernel(
    const float* __restrict__ rbf, const float* __restrict__ rhat,
    const int* __restrict__ mask,
    const float* __restrict__ a00, const float* __restrict__ a10,
    const float* __restrict__ a01, const float* __restrict__ a11,
    float* __restrict__ ws)
{
    const int jt = blockIdx.x, it = blockIdx.y;
    const int i0 = it * 16, j0 = jt * 16;
    const int tid = threadIdx.x;
    const int wid = tid >> 5;
    const int lane = tid & 31;
    const int lrow = lane & 15;     // M row / N col for fragments
    const int lhalf = lane >> 4;    // K-half selector

    __shared__ float rbfL[256 * 16];   // [pair][16], pair = ii*16+jj
    __shared__ float HmL[256 * 32];    // masked SiLU hidden, one branch at a time
    __shared__ float mL[256];          // mask as float
    __shared__ float rhL[16 * 48];     // [ii][jj][x]
    __shared__ float redA[256];        // [ii][o]
    __shared__ float redV[768];        // [x][ii][g]

    for (int idx = tid; idx < 4096; idx += 128) {
        const int p = idx >> 4, r = idx & 15;
        const int ii = p >> 4, jj = p & 15;
        rbfL[idx] = rbf[((size_t)(i0 + ii) * NN + (j0 + jj)) * 16 + r];
    }
    for (int idx = tid; idx < 256; idx += 128) {
        const int ii = idx >> 4, jj = idx & 15;
        mL[idx] = (float)mask[(i0 + ii) * NN + (j0 + jj)];
    }
    for (int idx = tid; idx < 768; idx += 128) {
        const int ii = idx / 48, rem = idx % 48;
        rhL[idx] = rhat[(size_t)(i0 + ii) * (NN * 3) + j0 * 3 + rem];
    }
    for (int idx = tid; idx < 256; idx += 128) redA[idx] = 0.f;
    for (int idx = tid; idx < 768; idx += 128) redV[idx] = 0.f;
    __syncthreads();

    v8f accA = {0.f, 0.f, 0.f, 0.f, 0.f, 0.f, 0.f, 0.f};
    v8f accV0 = accA, accV1 = accA, accV2 = accA;

    const float* atab[4] = {a00, a10, a01, a11};
    const float* Gtab[4] = {ws + OFF_G00, ws + OFF_G10, ws + OFF_G01, ws + OFF_G11};

    for (int br = 0; br < 4; ++br) {
        // ---- stage 1: Hm = mask * SiLU(rbf @ A + a), 64 pairs per wave ----
        const float* av = atab[br];
        const float* Ap = ws + OFF_AP + br * 512;
        v2f bf[2][4];
        #pragma unroll
        for (int nt = 0; nt < 2; ++nt)
            #pragma unroll
            for (int k = 0; k < 4; ++k)
                bf[nt][k] = ldg2(Ap + (nt * 4 + k) * 64 + lane * 2);
        const float abias0 = av[lrow];
        const float abias1 = av[16 + lrow];

        for (int mt = 0; mt < 4; ++mt) {
            const int pb = (wid * 4 + mt) * 16;
            #pragma unroll
            for (int nt = 0; nt < 2; ++nt) {
                v8f c = {0.f, 0.f, 0.f, 0.f, 0.f, 0.f, 0.f, 0.f};
                #pragma unroll
                for (int k = 0; k < 4; ++k) {
                    const int k0 = k * 4;
                    v2f a = *(const v2f*)&rbfL[(pb + lrow) * 16 + k0 + 2 * lhalf];
                    c = wmma4(a, bf[nt][k], c);
                }
                const float ab = nt ? abias1 : abias0;
                #pragma unroll
                for (int r = 0; r < 8; ++r) {
                    const int p = pb + r + 8 * lhalf;
                    float v = c[r] + ab;
                    v = v / (1.0f + __expf(-v));       // SiLU
                    HmL[p * 32 + nt * 16 + lrow] = v * mL[p];
                }
            }
        }
        __syncthreads();

        // ---- stage B: contraction GEMMs, K split by jj across 4 waves ----
        const float* G = Gtab[br];
        const int jlo = wid * 4;
        if (br == 0) {                     // msg00 -> accA
            for (int jj = jlo; jj < jlo + 4; ++jj) {
                const float* Gj = G + (size_t)(j0 + jj) * 512;
                #pragma unroll
                for (int k = 0; k < 8; ++k) {
                    v2f a = *(const v2f*)&HmL[(lrow * 16 + jj) * 32 + k * 4 + 2 * lhalf];
                    v2f b = ldg2(Gj + k * 64 + lane * 2);
                    accA = wmma4(a, b, accA);
                }
            }
        } else if (br == 1) {              // msg10 -> accA (r_hat folded into A)
            for (int jj = jlo; jj < jlo + 4; ++jj) {
                #pragma unroll
                for (int x = 0; x < 3; ++x) {
                    const float rv = rhL[lrow * 48 + jj * 3 + x];
                    const float* Gj = G + (size_t)(j0 + jj) * 1536 + x * 512;
                    #pragma unroll
                    for (int k = 0; k < 8; ++k) {
                        v2f a = *(const v2f*)&HmL[(lrow * 16 + jj) * 32 + k * 4 + 2 * lhalf];
                        a.x *= rv; a.y *= rv;
                        v2f b = ldg2(Gj + k * 64 + lane * 2);
                        accA = wmma4(a, b, accA);
                    }
                }
            }
        } else if (br == 2) {              // msg01 -> accV[c], shared B frag
            for (int jj = jlo; jj < jlo + 4; ++jj) {
                const float r0 = rhL[lrow * 48 + jj * 3 + 0];
                const float r1 = rhL[lrow * 48 + jj * 3 + 1];
                const float r2 = rhL[lrow * 48 + jj * 3 + 2];
                const float* Gj = G + (size_t)(j0 + jj) * 512;
                #pragma unroll
                for (int k = 0; k < 8; ++k) {
                    v2f a = *(const v2f*)&HmL[(lrow * 16 + jj) * 32 + k * 4 + 2 * lhalf];
                    v2f b = ldg2(Gj + k * 64 + lane * 2);
                    v2f a0 = {a.x * r0, a.y * r0};
                    v2f a1 = {a.x * r1, a.y * r1};
                    v2f a2 = {a.x * r2, a.y * r2};
                    accV0 = wmma4(a0, b, accV0);
                    accV1 = wmma4(a1, b, accV1);
                    accV2 = wmma4(a2, b, accV2);
                }
            }
        } else {                           // msg11 -> accV[x], shared A frag
            for (int jj = jlo; jj < jlo + 4; ++jj) {
                const float* Gj = G + (size_t)(j0 + jj) * 1536;
                #pragma unroll
                for (int k = 0; k < 8; ++k) {
                    v2f a = *(const v2f*)&HmL[(lrow * 16 + jj) * 32 + k * 4 + 2 * lhalf];
                    v2f b0 = ldg2(Gj + 0 * 512 + k * 64 + lane * 2);
                    v2f b1 = ldg2(Gj + 1 * 512 + k * 64 + lane * 2);
                    v2f b2 = ldg2(Gj + 2 * 512 + k * 64 + lane * 2);
                    accV0 = wmma4(a, b0, accV0);
                    accV1 = wmma4(a, b1, accV1);
                    accV2 = wmma4(a, b2, accV2);
                }
            }
        }
        __syncthreads();
    }

    // ---- cross-wave reduction of C tiles (deterministic staged adds) ----
    for (int w = 0; w < 4; ++w) {
        if (wid == w) {
            #pragma unroll
            for (int r = 0; r < 8; ++r) {
                const int M = r + 8 * lhalf;
                redA[M * 16 + lrow]           += accA[r];
                redV[0 * 256 + M * 16 + lrow] += accV0[r];
                redV[1 * 256 + M * 16 + lrow] += accV1[r];
                redV[2 * 256 + M * 16 + lrow] += accV2[r];
            }
        }
        __syncthreads();
    }

    // ---- bias (fb) epilogue, scalar ----
    const float* fb00 = ws + OFF_FB00;
    const float* fb01 = ws + OFF_FB01;
    const float* fb10 = ws + OFF_FB10;
    const float* fb11 = ws + OFF_FB11;
    for (int idx = tid; idx < 256; idx += 128) {
        const int ii = idx >> 4, o = idx & 15;
        float s = 0.f;
        for (int jj = 0; jj < 16; ++jj) {
            const float mj = mL[ii * 16 + jj];
            if (mj == 0.f) continue;
            const int j = j0 + jj;
            float t = fb00[j * 16 + o];
            #pragma unroll
            for (int x = 0; x < 3; ++x)
                t += rhL[ii * 48 + jj * 3 + x] * fb10[j * 48 + x * 16 + o];
            s += mj * t;
        }
        redA[idx] += s;
    }
    for (int idx = tid; idx < 768; idx += 128) {
        const int x = idx >> 8, rem = idx & 255;
        const int ii = rem >> 4, g = rem & 15;
        float s = 0.f;
        for (int jj = 0; jj < 16; ++jj) {
            const float mj = mL[ii * 16 + jj];
            if (mj == 0.f) continue;
            const int j = j0 + jj;
            s += mj * (rhL[ii * 48 + jj * 3 + x] * fb01[j * 16 + g] +
                       fb11[j * 48 + x * 16 + g]);
        }
        redV[idx] += s;
    }
    __syncthreads();

    // ---- write per-jt partials ----
    float* msgAp = ws + OFF_MA + (size_t)jt * (NN * 16);
    float* msgVp = ws + OFF_MV + (size_t)jt * (NN * 48);
    for (int idx = tid; idx < 256; idx += 128) {
        const int ii = idx >> 4, o = idx & 15;
        msgAp[(i0 + ii) * 16 + o] = redA[idx];
    }
    for (int idx = tid; idx < 768; idx += 128) {
        const int x = idx >> 8, rem = idx & 255;
        const int ii = rem >> 4, g = rem & 15;
        msgVp[(i0 + ii) * 48 + x * 16 + g] = redV[idx];
    }
}

// ---------------------------------------------------------------------------
// Kernel 3: reduce j-tile partials, LayerNorm f0, norm-gated f1.
// ---------------------------------------------------------------------------
__global__ __launch_bounds__(64) void finalize_kernel(
    const float* __restrict__ ws,
    const float* __restrict__ g0, const float* __restrict__ be0,
    const float* __restrict__ g1, const float* __restrict__ be1,
    float* __restrict__ out)
{
    const int i = blockIdx.x;
    const int tid = threadIdx.x;
    __shared__ float sA[16], sV[48], sN[16], stats[4];

    if (tid < 16) {
        float s = 0.f;
        for (int jt = 0; jt < 24; ++jt)
            s += ws[OFF_MA + (size_t)jt * (NN * 16) + i * 16 + tid];
        sA[tid] = s;
    }
    if (tid < 48) {
        float s = 0.f;
        for (int jt = 0; jt < 24; ++jt)
            s += ws[OFF_MV + (size_t)jt * (NN * 48) + i * 48 + tid];
        sV[tid] = s;   // layout [x][g]: x*16+g
    }
    __syncthreads();
    if (tid < 16) {
        const float v0 = sV[tid], v1 = sV[16 + tid], v2 = sV[32 + tid];
        sN[tid] = fmaxf(sqrtf(v0 * v0 + v1 * v1 + v2 * v2), 1e-8f);
    }
    __syncthreads();
    if (tid == 0) {
        float m0 = 0.f, mN = 0.f;
        for (int k = 0; k < 16; ++k) { m0 += sA[k]; mN += sN[k]; }
        m0 *= (1.f / 16.f); mN *= (1.f / 16.f);
        float v0 = 0.f, vN = 0.f;
        for (int k = 0; k < 16; ++k) {
            const float d0 = sA[k] - m0, dN = sN[k] - mN;
            v0 += d0 * d0; vN += dN * dN;
        }
        v0 *= (1.f / 16.f); vN *= (1.f / 16.f);
        stats[0] = m0; stats[1] = rsqrtf(v0 + 1e-5f);
        stats[2] = mN; stats[3] = rsqrtf(vN + 1e-5f);
    }
    __syncthreads();
    if (tid < 16)
        out[i * 16 + tid] = (sA[tid] - stats[0]) * stats[1] * g0[tid] + be0[tid];
    if (tid < 48) {
        const int x = tid >> 4, g = tid & 15;
        const float scale = (sN[g] - stats[2]) * stats[3] * g1[g] + be1[g];
        out[NN * 16 + i * 48 + g * 3 + x] = sV[x * 16 + g] / sN[g] * scale;
    }
}

// ---------------------------------------------------------------------------
extern "C" void kernel_launch(void* const* d_in, const int* in_sizes, int n_in,
                              void* d_out, int out_size, void* d_ws, size_t ws_size,
                              hipStream_t stream) {
    (void)in_sizes; (void)n_in; (void)out_size; (void)ws_size;
    const float* f0   = (const float*)d_in[1];
    const float* f1   = (const float*)d_in[2];
    const float* rbf  = (const float*)d_in[3];
    const float* rhat = (const float*)d_in[4];
    const int*   mask = (const int*)d_in[5];
    const float* A00 = (const float*)d_in[6],  *a00 = (const float*)d_in[7];
    const float* B00 = (const float*)d_in[8],  *b00 = (const float*)d_in[9];
    const float* A10 = (const float*)d_in[10], *a10 = (const float*)d_in[11];
    const float* B10 = (const float*)d_in[12], *b10 = (const float*)d_in[13];
    const float* A01 = (const float*)d_in[14], *a01 = (const float*)d_in[15];
    const float* B01 = (const float*)d_in[16], *b01 = (const float*)d_in[17];
    const float* A11 = (const float*)d_in[18], *a11 = (const float*)d_in[19];
    const float* B11 = (const float*)d_in[20], *b11 = (const float*)d_in[21];
    const float* g0  = (const float*)d_in[22], *be0 = (const float*)d_in[23];
    const float* g1  = (const float*)d_in[24], *be1 = (const float*)d_in[25];
    float* wsf = (float*)d_ws;
    float* outf = (float*)d_out;

    precompute_kernel<<<NN, 256, 0, stream>>>(f0, f1, B00, b00, B10, b10,
                                              B01, b01, B11, b11,
                                              A00, A10, A01, A11, wsf);
    pair_kernel<<<dim3(24, 24), 128, 0, stream>>>(rbf, rhat, mask,
                                                  a00, a10, a01, a11, wsf);
    finalize_kernel<<<NN, 64, 0, stream>>>(wsf, g0, be0, g1, be1, outf);
}